// SelectSageModule_49426483642533
// MI455X (gfx1250) — compile-verified
//
#include <hip/hip_runtime.h>
#include <hip/hip_bf16.h>
#include <math.h>

// ---------------------------------------------------------------------------
// Problem constants (match reference)
// ---------------------------------------------------------------------------
#define DIM    128
#define NPER   2048
#define KSEL   1024

typedef float v2f __attribute__((ext_vector_type(2)));
typedef float v8f __attribute__((ext_vector_type(8)));

// ---------------------------------------------------------------------------
// Kernel 1: fused projection p[i] = x[i]·W_rel , attn[i] = x[i]·W_root + b_rel
// using V_WMMA_F32_16X16X4_F32.  Each wave computes 16 rows; K=128 in 32 steps.
// B fragment: 4x16 tile, col 0 = W_rel slice, col 1 = W_root slice, rest 0.
// Branchless B loads: lanes n>=2 read from a zeroed LDS column so EXEC never
// changes and all ds_load_b64 issue unconditionally (no saveexec pairs).
// ---------------------------------------------------------------------------
__global__ __launch_bounds__(256) void proj_wmma_kernel(
    const float* __restrict__ x,
    const float* __restrict__ Wrel,
    const float* __restrict__ Wroot,
    const float* __restrict__ brel,
    float* __restrict__ p,
    float* __restrict__ attn,
    int nrows)
{
    // [0..127]=W_rel, [128..255]=W_root, [256..383]=0 (dead-column source)
    __shared__ float sW[3 * DIM];
    for (int i = threadIdx.x; i < 3 * DIM; i += blockDim.x)
        sW[i] = (i < DIM) ? Wrel[i] : (i < 2 * DIM) ? Wroot[i - DIM] : 0.0f;
    __syncthreads();

    const int lane    = threadIdx.x & 31;
    const int wave    = threadIdx.x >> 5;
    const int waveId  = blockIdx.x * (blockDim.x >> 5) + wave;
    const int rowBase = waveId << 4;          // 16 rows per wave
    if (rowBase + 16 > nrows) return;         // uniform per wave -> EXEC stays full

    const int n   = lane & 15;                // A: row within tile / C: column N
    const int hi  = lane >> 4;                // K sub-group select
    const float* xrow = x + (size_t)(rowBase + n) * DIM;

    // Column source: n==0 -> W_rel, n==1 -> W_root, n>=2 -> zero column.
    const int colSel = (n < 2) ? n : 2;
    const float* wcol = sW + colSel * DIM;

    v8f c = {};
    #pragma unroll
    for (int kk = 0; kk < 32; ++kk) {
        const int k0 = kk * 4 + hi * 2;       // A/B VGPR0 -> K = 2*hi, VGPR1 -> K = 2*hi+1
        v2f a;
        a.x = xrow[k0];
        a.y = xrow[k0 + 1];
        v2f b;
        b.x = wcol[k0];
        b.y = wcol[k0 + 1];
        c = __builtin_amdgcn_wmma_f32_16x16x4_f32(
                /*neg_a=*/false, a, /*neg_b=*/false, b,
                /*c_mod=*/(short)0, c, /*reuse_a=*/false, /*reuse_b=*/false);
    }

    // C layout: VGPR r, lane L -> M = r + 8*(L>>4), N = L & 15
    const float bb = brel[0];
    if (n == 0) {
        #pragma unroll
        for (int r = 0; r < 8; ++r)
            p[rowBase + r + 8 * hi] = c[r];
    } else if (n == 1) {
        #pragma unroll
        for (int r = 0; r < 8; ++r)
            attn[rowBase + r + 8 * hi] = c[r] + bb;
    }
}

// ---------------------------------------------------------------------------
// Kernel 2: edge scatter  attn[dst[e]] += p[src[e]]  (scalar per edge).
// int4-vectorized index loads; non-returning f32 atomics.
// ---------------------------------------------------------------------------
__global__ __launch_bounds__(256) void edge_scatter_kernel(
    const int* __restrict__ src,
    const int* __restrict__ dst,
    const float* __restrict__ p,
    float* __restrict__ attn,
    int E)
{
    const int stride = gridDim.x * blockDim.x;
    const int tid    = blockIdx.x * blockDim.x + threadIdx.x;
    const int E4     = E >> 2;
    const int4* s4   = (const int4*)src;
    const int4* d4   = (const int4*)dst;

    for (int e = tid; e < E4; e += stride) {
        const int4 s = s4[e];
        const int4 d = d4[e];
        atomicAdd(&attn[d.x], p[s.x]);
        atomicAdd(&attn[d.y], p[s.y]);
        atomicAdd(&attn[d.z], p[s.z]);
        atomicAdd(&attn[d.w], p[s.w]);
    }
    for (int e = (E4 << 2) + tid; e < E; e += stride)
        atomicAdd(&attn[dst[e]], p[src[e]]);
}

// ---------------------------------------------------------------------------
// Kernel 3: per-graph score + top-k.  One 1024-thread block per graph;
// bitonic sort of 2048 (value desc, index asc on ties) entirely in LDS.
// ---------------------------------------------------------------------------
__global__ __launch_bounds__(1024) void topk_kernel(
    const float* __restrict__ attn,
    const float* __restrict__ selw,
    int* __restrict__ out_idx,
    float* __restrict__ out_val)
{
    __shared__ float sv[NPER];
    __shared__ int   si[NPER];

    const int g   = blockIdx.x;
    const int tid = threadIdx.x;
    const float w = selw[0];
    const float s = w / sqrtf(w * w);         // == sign(w); matches reference math
    const float* a = attn + (size_t)g * NPER;

    for (int i = tid; i < NPER; i += 1024) {
        sv[i] = tanhf(a[i] * s);
        si[i] = i;
    }
    __syncthreads();

    // Bitonic sort, final order: descending value, ascending index on ties.
    for (int k = 2; k <= NPER; k <<= 1) {
        for (int j = k >> 1; j > 0; j >>= 1) {
            const int l = ((tid & ~(j - 1)) << 1) | (tid & (j - 1));
            const int m = l | j;
            const bool desc = ((l & k) == 0);
            const float vl = sv[l], vm = sv[m];
            const int   il = si[l], im = si[m];
            // "l ranks before m" in descending order
            const bool inOrder = (vl > vm) || (vl == vm && il < im);
            const bool doSwap  = desc ? !inOrder : inOrder;
            if (doSwap) {
                sv[l] = vm; sv[m] = vl;
                si[l] = im; si[m] = il;
            }
            __syncthreads();
        }
    }

    for (int i = tid; i < KSEL; i += 1024) {
        out_idx[(size_t)g * KSEL + i] = si[i] + g * NPER;   // global node id
        out_val[(size_t)g * KSEL + i] = sv[i];
    }
}

// ---------------------------------------------------------------------------
// Host-side launch
// ---------------------------------------------------------------------------
extern "C" void kernel_launch(void* const* d_in, const int* in_sizes, int n_in,
                              void* d_out, int out_size, void* d_ws, size_t ws_size,
                              hipStream_t stream)
{
    const float* x     = (const float*)d_in[0];   // [N, 128]
    const int*   ei    = (const int*)  d_in[1];   // [2, E]
    // d_in[2] = batch (unused: graphs are contiguous, equal size)
    const float* Wrel  = (const float*)d_in[3];   // [128]
    const float* brel  = (const float*)d_in[4];   // [1]
    const float* Wroot = (const float*)d_in[5];   // [128]
    const float* selw  = (const float*)d_in[6];   // [1]

    const int N = in_sizes[0] / DIM;
    const int E = in_sizes[1] / 2;
    const int B = N / NPER;

    float* p    = (float*)d_ws;          // [N]
    float* attn = p + N;                 // [N]

    int*   out_idx = (int*)d_out;                        // first B*K (int32 bits)
    float* out_val = (float*)d_out + (size_t)B * KSEL;   // next  B*K (f32)

    // 16 rows/wave, 8 waves/block -> 128 rows/block
    const int projBlocks = (N + 127) / 128;
    proj_wmma_kernel<<<projBlocks, 256, 0, stream>>>(x, Wrel, Wroot, brel, p, attn, N);

    edge_scatter_kernel<<<2048, 256, 0, stream>>>(ei, ei + E, p, attn, E);

    topk_kernel<<<B, 1024, 0, stream>>>(attn, selw, out_idx, out_val);
}